// GraphPooler_65566970740941
// MI455X (gfx1250) — compile-verified
//
#include <hip/hip_runtime.h>
#include <hip/hip_bf16.h>

typedef __attribute__((ext_vector_type(2))) float v2f;
typedef __attribute__((ext_vector_type(4))) float v4f;
typedef __attribute__((ext_vector_type(8))) float v8f;

#define D_FEAT 128

// ---------------------------------------------------------------------------
// Kernel 1: exclusive prefix sum of graph_size (int64) -> int32 node offsets.
// Single block; G ~ 10000 so chunked scan is microseconds.
// ---------------------------------------------------------------------------
__global__ __launch_bounds__(256)
void gp_scan_offsets(const long long* __restrict__ gs, int* __restrict__ offsets, int G) {
    __shared__ int lsum[256];
    const int tid = threadIdx.x;
    const int per = (G + 255) / 256;
    const int start = tid * per;
    const int end   = (start + per < G) ? (start + per) : G;
    int s = 0;
    for (int i = start; i < end; ++i) s += (int)gs[i];
    lsum[tid] = s;
    __syncthreads();
    // inclusive Hillis-Steele scan over 256 partial sums
    for (int off = 1; off < 256; off <<= 1) {
        int v = (tid >= off) ? lsum[tid - off] : 0;
        __syncthreads();
        lsum[tid] += v;
        __syncthreads();
    }
    int run = (tid == 0) ? 0 : lsum[tid - 1];   // exclusive base for this chunk
    for (int i = start; i < end; ++i) {
        offsets[i] = run;
        run += (int)gs[i];
    }
}

// ---------------------------------------------------------------------------
// Kernel 2: per-graph mean+max pooling. One workgroup (8 waves) per graph.
// Each wave loads one full 128-float row per iteration as 32 lanes x float4,
// using non-temporal loads (1 GB streamed once; keep it out of L2).
// pooled[g][0:128]   = mean,  pooled[g][128:256] = max
// ---------------------------------------------------------------------------
__global__ __launch_bounds__(256)
void gp_pool(const float* __restrict__ feats,
             const long long* __restrict__ gs,
             const int* __restrict__ offsets,
             float* __restrict__ pooled) {
    const int g    = blockIdx.x;
    const int tid  = threadIdx.x;
    const int lane = tid & 31;
    const int wave = tid >> 5;

    const int base = offsets[g];
    const int cnt  = (int)gs[g];

    v4f sum = {0.f, 0.f, 0.f, 0.f};
    v4f mx  = {-__builtin_inff(), -__builtin_inff(), -__builtin_inff(), -__builtin_inff()};

    for (int r = wave; r < cnt; r += 8) {
        const v4f* p = (const v4f*)(feats + (size_t)(base + r) * D_FEAT) + lane;
        v4f v = __builtin_nontemporal_load(p);
        sum.x += v.x; sum.y += v.y; sum.z += v.z; sum.w += v.w;
        mx.x = fmaxf(mx.x, v.x); mx.y = fmaxf(mx.y, v.y);
        mx.z = fmaxf(mx.z, v.z); mx.w = fmaxf(mx.w, v.w);
    }

    __shared__ float s_sum[8][D_FEAT];
    __shared__ float s_max[8][D_FEAT];
    s_sum[wave][lane * 4 + 0] = sum.x;  s_max[wave][lane * 4 + 0] = mx.x;
    s_sum[wave][lane * 4 + 1] = sum.y;  s_max[wave][lane * 4 + 1] = mx.y;
    s_sum[wave][lane * 4 + 2] = sum.z;  s_max[wave][lane * 4 + 2] = mx.z;
    s_sum[wave][lane * 4 + 3] = sum.w;  s_max[wave][lane * 4 + 3] = mx.w;
    __syncthreads();

    if (tid < D_FEAT) {
        float s = s_sum[0][tid];
        float m = s_max[0][tid];
        #pragma unroll
        for (int w = 1; w < 8; ++w) {
            s += s_sum[w][tid];
            m = fmaxf(m, s_max[w][tid]);
        }
        const float inv = 1.0f / (float)cnt;
        pooled[(size_t)g * 256 + tid]          = s * inv;
        pooled[(size_t)g * 256 + D_FEAT + tid] = m;
    }
}

// ---------------------------------------------------------------------------
// Kernel 3: fused 3-layer MLP with f32 WMMA (V_WMMA_F32_16X16X4_F32).
// Block = 512 threads = 16 waves, handles 16 graphs (rows).
//   Layer0: 256->256, wave t owns output cols [16t,16t+16), K-loop step 4.
//   Layer1: 256->128, waves 0..7.
//   Layer2: 128->1 dot + sigmoid, threads 0..15.
// A fragment (16x4 f32): lane l -> row m=l&15, K pair base 2*(l>>4).
// B fragment (4x16 f32): lane l -> col n=l&15, same K pair base.
// C/D (16x16 f32, 8 VGPRs): vgpr r -> row r + 8*(l>>4), col l&15.
// ---------------------------------------------------------------------------
__global__ __launch_bounds__(512)
void gp_mlp(const float* __restrict__ pooled,
            const float* __restrict__ W0, const float* __restrict__ b0,
            const float* __restrict__ W1, const float* __restrict__ b1,
            const float* __restrict__ W2, const float* __restrict__ b2,
            float* __restrict__ out, int G) {
    __shared__ float s_in[16][257];   // +1 pad: A-fragment column gathers hit distinct banks
    __shared__ float s_h0[16][257];
    __shared__ float s_h1[16][129];

    const int g0   = blockIdx.x * 16;
    const int tid  = threadIdx.x;
    const int lane = tid & 31;
    const int wave = tid >> 5;
    const int m    = lane & 15;      // A row / B and C column within tile
    const int half = lane >> 4;      // K-pair / C-row-half selector
    const int kb   = 2 * half;

    // Load pooled tile [16 x 256]
    for (int i = tid; i < 16 * 256; i += 512) {
        const int row = i >> 8, col = i & 255;
        const int gg = g0 + row;
        s_in[row][col] = (gg < G) ? pooled[(size_t)gg * 256 + col] : 0.0f;
    }
    __syncthreads();

    // ---- Layer 0: [16x256] @ [256x256] + b0, relu ----
    {
        const int t = wave;              // N-tile 0..15
        const int n = t * 16 + m;        // global output column
        v8f acc = {0.f, 0.f, 0.f, 0.f, 0.f, 0.f, 0.f, 0.f};
        for (int k0 = 0; k0 < 256; k0 += 4) {
            v2f a;
            a.x = s_in[m][k0 + kb];
            a.y = s_in[m][k0 + kb + 1];
            v2f b;
            b.x = W0[(size_t)(k0 + kb)     * 256 + n];
            b.y = W0[(size_t)(k0 + kb + 1) * 256 + n];
            acc = __builtin_amdgcn_wmma_f32_16x16x4_f32(
                false, a, false, b, (short)0, acc, false, false);
        }
        const float bias = b0[n];
        #pragma unroll
        for (int r = 0; r < 8; ++r) {
            float v = acc[r] + bias;
            s_h0[r + half * 8][n] = v > 0.f ? v : 0.f;
        }
    }
    __syncthreads();

    // ---- Layer 1: [16x256] @ [256x128] + b1, relu ----
    if (wave < 8) {                      // wave-uniform branch: EXEC stays all-ones
        const int t = wave;
        const int n = t * 16 + m;
        v8f acc = {0.f, 0.f, 0.f, 0.f, 0.f, 0.f, 0.f, 0.f};
        for (int k0 = 0; k0 < 256; k0 += 4) {
            v2f a;
            a.x = s_h0[m][k0 + kb];
            a.y = s_h0[m][k0 + kb + 1];
            v2f b;
            b.x = W1[(size_t)(k0 + kb)     * 128 + n];
            b.y = W1[(size_t)(k0 + kb + 1) * 128 + n];
            acc = __builtin_amdgcn_wmma_f32_16x16x4_f32(
                false, a, false, b, (short)0, acc, false, false);
        }
        const float bias = b1[n];
        #pragma unroll
        for (int r = 0; r < 8; ++r) {
            float v = acc[r] + bias;
            s_h1[r + half * 8][n] = v > 0.f ? v : 0.f;
        }
    }
    __syncthreads();

    // ---- Layer 2: [16x128] @ [128x1] + b2, sigmoid ----
    if (tid < 16) {
        const int gg = g0 + tid;
        if (gg < G) {
            float s = 0.f;
            #pragma unroll 4
            for (int d = 0; d < 128; ++d) s += s_h1[tid][d] * W2[d];
            s += b2[0];
            out[gg] = 1.0f / (1.0f + __expf(-s));
        }
    }
}

// ---------------------------------------------------------------------------
extern "C" void kernel_launch(void* const* d_in, const int* in_sizes, int n_in,
                              void* d_out, int out_size, void* d_ws, size_t ws_size,
                              hipStream_t stream) {
    const float*     self_feats = (const float*)d_in[0];
    const long long* graph_size = (const long long*)d_in[1];
    const float*     W0 = (const float*)d_in[2];
    const float*     b0 = (const float*)d_in[3];
    const float*     W1 = (const float*)d_in[4];
    const float*     b1 = (const float*)d_in[5];
    const float*     W2 = (const float*)d_in[6];
    const float*     b2 = (const float*)d_in[7];
    float* out = (float*)d_out;

    const int G = in_sizes[1];

    // workspace: [ int offsets[G] | pad to 256B | float pooled[G*256] ]
    int*   offsets = (int*)d_ws;
    size_t off_bytes = (((size_t)G * sizeof(int)) + 255) & ~(size_t)255;
    float* pooled  = (float*)((char*)d_ws + off_bytes);

    gp_scan_offsets<<<1, 256, 0, stream>>>(graph_size, offsets, G);
    gp_pool<<<G, 256, 0, stream>>>(self_feats, graph_size, offsets, pooled);
    gp_mlp<<<(G + 15) / 16, 512, 0, stream>>>(pooled, W0, b0, W1, b1, W2, b2, out, G);
}